// GAT_51075751084502
// MI455X (gfx1250) — compile-verified
//
#include <hip/hip_runtime.h>
#include <math.h>

// ---------------- problem constants (match reference) ----------------
static constexpr int N_NODES = 50000;
static constexpr int N_EDGES = 500000;
static constexpr int T_SNAP  = 2;
static constexpr int NETY    = 10;
static constexpr int DE      = 16;
static constexpr int NH      = 4;
static constexpr int F0 = 64, F1 = 128, F2 = 64;
static constexpr int C0 = F1 / NH;   // 32
static constexpr int C1 = F2 / NH;   // 16
static constexpr int E2 = N_EDGES + N_NODES;   // edges + self loops

typedef float v2f __attribute__((ext_vector_type(2)));
typedef float v8f __attribute__((ext_vector_type(8)));
typedef int   v4i __attribute__((ext_vector_type(4)));

// CDNA5 async global->LDS path (guarded: falls back to ds-store staging if the
// toolchain does not expose the builtins)
#if defined(__HIP_DEVICE_COMPILE__) && \
    __has_builtin(__builtin_amdgcn_global_load_async_to_lds_b128) && \
    __has_builtin(__builtin_amdgcn_s_wait_asynccnt)
#define GAT_ASYNC_LDS 1
// typed address-space pointers matching the builtin prototype:
//   (v4i addrspace(1)*, v4i addrspace(3)*, imm offset, imm cpol)
typedef __attribute__((address_space(1))) v4i as1_v4i_t;
typedef __attribute__((address_space(3))) v4i as3_v4i_t;
#else
#define GAT_ASYNC_LDS 0
#endif

static inline int cdiv(int a, int b) { return (a + b - 1) / b; }

// monotonic float <-> uint encoding so atomicMax(u32) == float max
__device__ __forceinline__ unsigned encf(float f) {
  unsigned u = __float_as_uint(f);
  return (u & 0x80000000u) ? ~u : (u | 0x80000000u);
}
__device__ __forceinline__ float decf(unsigned u) {
  return (u & 0x80000000u) ? __uint_as_float(u & 0x7FFFFFFFu) : __uint_as_float(~u);
}

// ---------------- trivial init kernels ----------------
__global__ void gat_fill_f32(float* __restrict__ p, int n, float v) {
  int i = blockIdx.x * blockDim.x + threadIdx.x;
  if (i < n) p[i] = v;
}
__global__ void gat_fill_u32(unsigned* __restrict__ p, int n, unsigned v) {
  int i = blockIdx.x * blockDim.x + threadIdx.x;
  if (i < n) p[i] = v;
}

// ---------------- self-loop edge_attr fill (scatter-mean of emb[etype] by dst) ----------------
__global__ void gat_accum_fill(const int* __restrict__ dst, const int* __restrict__ ety,
                               const float* __restrict__ emb,
                               float* __restrict__ fillb, float* __restrict__ deg, int E) {
  int i = blockIdx.x * blockDim.x + threadIdx.x;
  if (i >= E * DE) return;
  int e = i >> 4, j = i & 15;
  int d = dst[e], et = ety[e];
  atomicAdd(&fillb[d * DE + j], emb[et * DE + j]);
  if (j == 0) atomicAdd(&deg[d], 1.0f);
}
__global__ void gat_fin_fill(float* __restrict__ fillb, const float* __restrict__ deg, int n) {
  int i = blockIdx.x * blockDim.x + threadIdx.x;
  if (i >= n) return;
  fillb[i] = fillb[i] / fmaxf(deg[i >> 4], 1.0f);
}

// ---------------- WMMA fp32 GEMM: Y[nRows,NCOL] = X[nRows,K] @ W[K,NCOL] ----------------
// Block = 128 threads (4 waves) computes a 64x64 tile: each wave owns one 16-col
// slice and 4 row-tiles (4 accumulators), so one B fragment feeds 4 WMMAs.
// X panel staged in LDS (async global->LDS b128 on CDNA5), W streamed (L2-resident).
template <int K, int NCOL>
__global__ __launch_bounds__(128) void gat_gemm_wmma(const float* __restrict__ X,
                                                     const float* __restrict__ W,
                                                     float* __restrict__ Y,
                                                     int nRows) {
  constexpr int KP = K + 4;                  // pad 4 floats: keeps 16B align, kills bank conflicts
  __shared__ float As[64 * KP];
  const int row0 = blockIdx.x * 64;
  const int tid  = threadIdx.x;
  const int wave = tid >> 5;
  const int lane = tid & 31;

  // stage 64 x K panel of X into LDS, 16B per lane per step
  constexpr int NV4 = 16 * K;                // (64*K)/4 float4 elements
  for (int i = tid; i < NV4; i += 128) {
    int r = i / (K / 4);
    int k = (i - r * (K / 4)) * 4;
    int gr = row0 + r;
    if (gr > nRows - 1) gr = nRows - 1;      // clamp: guarded out at store
    const float* src = X + (size_t)gr * K + k;
    float* dst = &As[r * KP + k];
#if GAT_ASYNC_LDS
    __builtin_amdgcn_global_load_async_to_lds_b128(
        (as1_v4i_t*)(__attribute__((address_space(1))) void*)src,
        (as3_v4i_t*)(__attribute__((address_space(3))) void*)dst, 0, 0);
#else
    *(float4*)dst = *(const float4*)src;
#endif
  }
#if GAT_ASYNC_LDS
  __builtin_amdgcn_s_wait_asynccnt(0);
#endif
  __syncthreads();

  const int colb = blockIdx.y * 64 + wave * 16;
  const int g  = lane >> 4;                  // half-wave group: selects K pair
  const int mn = lane & 15;                  // row (A) / col (B/D) index in tile

  v8f acc[4];
#pragma unroll
  for (int rt = 0; rt < 4; ++rt) acc[rt] = v8f{0.f, 0.f, 0.f, 0.f, 0.f, 0.f, 0.f, 0.f};

  for (int kk = 0; kk < K; kk += 4) {
    const int k0 = kk + g * 2;
    v2f b;
    b.x = W[k0 * NCOL + colb + mn];
    b.y = W[(k0 + 1) * NCOL + colb + mn];
    if (kk + 8 < K)                          // gfx1250 prefetch path for next W tile
      __builtin_prefetch((const void*)(W + (kk + 8) * NCOL + colb + mn), 0, 1);
#pragma unroll
    for (int rt = 0; rt < 4; ++rt) {
      v2f a = *(const v2f*)&As[(rt * 16 + mn) * KP + k0];
      // D = A(16x4,f32) * B(4x16,f32) + C ; fp32 accumulate = reference numerics
      acc[rt] = __builtin_amdgcn_wmma_f32_16x16x4_f32(false, a, false, b, (short)0,
                                                      acc[rt], false, false);
    }
  }

  // D layout: acc[rt][v] = D[rt*16 + v + 8*g][mn]
#pragma unroll
  for (int rt = 0; rt < 4; ++rt) {
#pragma unroll
    for (int v = 0; v < 8; ++v) {
      int r = row0 + rt * 16 + v + 8 * g;
      if (r < nRows) Y[(size_t)r * NCOL + colb + mn] = acc[rt][v];
    }
  }
}

// ---------------- per-layer tiny precompute: q[h,d] and etab[et,h] ----------------
// q[h][d] = sum_c We[d, h*C+c] * att2[h,c];  etab[et][h] = sum_d emb[et,d]*q[h,d]
__global__ void gat_precompute(const float* __restrict__ We, const float* __restrict__ att,
                               const float* __restrict__ emb,
                               float* __restrict__ q, float* __restrict__ etab, int C) {
  __shared__ float qs[NH * DE];
  int tid = threadIdx.x;
  int HC = NH * C;
  if (tid < NH * DE) {
    int h = tid / DE, d = tid % DE;
    float s = 0.f;
    for (int c = 0; c < C; ++c)
      s += We[(size_t)d * HC + h * C + c] * att[(2 * NH + h) * C + c];
    qs[tid] = s;
    q[tid] = s;
  }
  __syncthreads();
  if (tid < NETY * NH) {
    int et = tid / NH, h = tid % NH;
    float s = 0.f;
    for (int d = 0; d < DE; ++d) s += emb[et * DE + d] * qs[h * DE + d];
    etab[tid] = s;
  }
}

// ---------------- per-(node,head) attention scalars ----------------
template <int C>
__global__ void gat_node_scalars(const float* __restrict__ xw, const float* __restrict__ att,
                                 const float* __restrict__ fillb, const float* __restrict__ q,
                                 float* __restrict__ asrc, float* __restrict__ adst,
                                 float* __restrict__ aself) {
  constexpr int HC = NH * C;
  int i = blockIdx.x * blockDim.x + threadIdx.x;
  if (i >= N_NODES * NH) return;
  int n = i / NH, h = i % NH;
  const float* xr = xw + (size_t)n * HC + h * C;
  float s0 = 0.f, s1 = 0.f;
#pragma unroll
  for (int c = 0; c < C; ++c) {
    s0 += xr[c] * att[(0 * NH + h) * C + c];
    s1 += xr[c] * att[(1 * NH + h) * C + c];
  }
  float s2 = 0.f;
#pragma unroll
  for (int d = 0; d < DE; ++d) s2 += fillb[(size_t)n * DE + d] * q[h * DE + d];
  asrc[i] = s0; adst[i] = s1; aself[i] = s2;
}

// ---------------- pass A: logits (leaky relu) + segment max ----------------
__global__ void gat_edge_logits(const int* __restrict__ src, const int* __restrict__ dst,
                                const int* __restrict__ ety,
                                const float* __restrict__ asrc, const float* __restrict__ adst,
                                const float* __restrict__ aself, const float* __restrict__ etab,
                                float* __restrict__ logits, unsigned* __restrict__ menc) {
  int e = blockIdx.x * blockDim.x + threadIdx.x;
  if (e >= E2) return;
  int s, d;
  float term[NH];
  if (e < N_EDGES) {
    s = src[e]; d = dst[e];
    int et = ety[e];
#pragma unroll
    for (int h = 0; h < NH; ++h) term[h] = etab[et * NH + h];
  } else {
    s = d = e - N_EDGES;
#pragma unroll
    for (int h = 0; h < NH; ++h) term[h] = aself[(size_t)(e - N_EDGES) * NH + h];
  }
#pragma unroll
  for (int h = 0; h < NH; ++h) {
    float l = asrc[(size_t)s * NH + h] + adst[(size_t)d * NH + h] + term[h];
    l = l > 0.f ? l : 0.2f * l;                  // leaky_relu(0.2)
    logits[(size_t)e * NH + h] = l;
    atomicMax(&menc[d * NH + h], encf(l));
  }
}

// ---------------- pass B: exp(l - max) + segment sum ----------------
__global__ void gat_edge_exp(const int* __restrict__ dst, float* __restrict__ logits,
                             const unsigned* __restrict__ menc, float* __restrict__ ssum) {
  int i = blockIdx.x * blockDim.x + threadIdx.x;
  if (i >= E2 * NH) return;
  int e = i >> 2, h = i & 3;
  int d = (e < N_EDGES) ? dst[e] : (e - N_EDGES);
  float w = expf(logits[i] - decf(menc[d * NH + h]));
  logits[i] = w;
  atomicAdd(&ssum[d * NH + h], w);
}

// ---------------- pass C: weighted scatter-sum of xw[src] into out[dst] ----------------
template <int C>
__global__ void gat_scatter(const int* __restrict__ src, const int* __restrict__ dst,
                            const float* __restrict__ logits, const float* __restrict__ ssum,
                            const float* __restrict__ xw, float* __restrict__ outb,
                            int rowStride) {
  constexpr int HC = NH * C;
  int i = blockIdx.x * blockDim.x + threadIdx.x;
  if (i >= E2 * HC) return;
  int e = i / HC, j = i - e * HC;
  int h = j / C;
  int d, s;
  if (e < N_EDGES) { d = dst[e]; s = src[e]; } else { d = s = e - N_EDGES; }
  float coef = logits[(size_t)e * NH + h] / ssum[d * NH + h];
  atomicAdd(&outb[(size_t)d * rowStride + j], xw[(size_t)s * HC + j] * coef);
}

// ---------------- epilogues ----------------
__global__ void gat_bias_relu(float* __restrict__ buf, const float* __restrict__ b, int n, int F) {
  int i = blockIdx.x * blockDim.x + threadIdx.x;
  if (i >= n) return;
  buf[i] = fmaxf(buf[i] + b[i % F], 0.f);
}
__global__ void gat_init_out_bias(float* __restrict__ base, const float* __restrict__ b,
                                  int rowStride, int F, int n) {
  int i = blockIdx.x * blockDim.x + threadIdx.x;
  if (i >= n) return;
  int r = i / F, f = i - r * F;
  base[(size_t)r * rowStride + f] = b[f];
}

// ---------------- host launcher ----------------
extern "C" void kernel_launch(void* const* d_in, const int* in_sizes, int n_in,
                              void* d_out, int out_size, void* d_ws, size_t ws_size,
                              hipStream_t stream) {
  const float* x     = (const float*)d_in[0];   // [T,N,F0]
  const int*   eidx  = (const int*)  d_in[1];   // [T,2,E]
  const int*   etyp  = (const int*)  d_in[2];   // [T,E]
  const float* emb   = (const float*)d_in[3];   // [NET,DE]
  const float* W0    = (const float*)d_in[4];
  const float* We0   = (const float*)d_in[5];
  const float* att0  = (const float*)d_in[6];
  const float* b0    = (const float*)d_in[7];
  const float* W1    = (const float*)d_in[8];
  const float* We1   = (const float*)d_in[9];
  const float* att1  = (const float*)d_in[10];
  const float* b1    = (const float*)d_in[11];
  float* out = (float*)d_out;                   // [N,T,F2]

  // carve workspace
  char* w = (char*)d_ws;
  auto carve = [&](size_t bytes) -> void* {
    void* p = (void*)w;
    w += (bytes + 255) & ~(size_t)255;
    return p;
  };
  float*    fillb = (float*)   carve((size_t)N_NODES * DE * 4);
  float*    deg   = (float*)   carve((size_t)N_NODES * 4);
  float*    xw    = (float*)   carve((size_t)N_NODES * F1 * 4);   // reused (layer1 uses N*F2)
  float*    acc0  = (float*)   carve((size_t)N_NODES * F1 * 4);   // layer0 accum, then h1
  float*    asrc  = (float*)   carve((size_t)N_NODES * NH * 4);
  float*    adst  = (float*)   carve((size_t)N_NODES * NH * 4);
  float*    aself = (float*)   carve((size_t)N_NODES * NH * 4);
  unsigned* menc  = (unsigned*)carve((size_t)N_NODES * NH * 4);
  float*    ssum  = (float*)   carve((size_t)N_NODES * NH * 4);
  float*    wexp  = (float*)   carve((size_t)E2 * NH * 4);        // logits then exp weights
  float*    qbuf  = (float*)   carve((size_t)NH * DE * 4);
  float*    etab  = (float*)   carve((size_t)NETY * NH * 4);

  const int B = 256;

  for (int t = 0; t < T_SNAP; ++t) {
    const float* xt   = x    + (size_t)t * N_NODES * F0;
    const int*   srcP = eidx + (size_t)t * 2 * N_EDGES;
    const int*   dstP = srcP + N_EDGES;
    const int*   etP  = etyp + (size_t)t * N_EDGES;

    // self-loop edge-attr fill (scatter mean)
    gat_fill_f32<<<cdiv(N_NODES * DE, B), B, 0, stream>>>(fillb, N_NODES * DE, 0.f);
    gat_fill_f32<<<cdiv(N_NODES, B), B, 0, stream>>>(deg, N_NODES, 0.f);
    gat_accum_fill<<<cdiv(N_EDGES * DE, B), B, 0, stream>>>(dstP, etP, emb, fillb, deg, N_EDGES);
    gat_fin_fill<<<cdiv(N_NODES * DE, B), B, 0, stream>>>(fillb, deg, N_NODES * DE);

    // ---------------- layer 0 ----------------
    gat_precompute<<<1, 64, 0, stream>>>(We0, att0, emb, qbuf, etab, C0);
    gat_gemm_wmma<F0, F1><<<dim3(cdiv(N_NODES, 64), F1 / 64), 128, 0, stream>>>(xt, W0, xw,
                                                                                N_NODES);
    gat_node_scalars<C0><<<cdiv(N_NODES * NH, B), B, 0, stream>>>(xw, att0, fillb, qbuf,
                                                                  asrc, adst, aself);
    gat_fill_u32<<<cdiv(N_NODES * NH, B), B, 0, stream>>>(menc, N_NODES * NH, 0u);
    gat_fill_f32<<<cdiv(N_NODES * NH, B), B, 0, stream>>>(ssum, N_NODES * NH, 0.f);
    gat_fill_f32<<<cdiv(N_NODES * F1, B), B, 0, stream>>>(acc0, N_NODES * F1, 0.f);
    gat_edge_logits<<<cdiv(E2, B), B, 0, stream>>>(srcP, dstP, etP, asrc, adst, aself,
                                                   etab, wexp, menc);
    gat_edge_exp<<<cdiv(E2 * NH, B), B, 0, stream>>>(dstP, wexp, menc, ssum);
    gat_scatter<C0><<<cdiv(E2 * F1, B), B, 0, stream>>>(srcP, dstP, wexp, ssum, xw, acc0, F1);
    gat_bias_relu<<<cdiv(N_NODES * F1, B), B, 0, stream>>>(acc0, b0, N_NODES * F1, F1);
    // acc0 now holds h1 [N,F1]

    // ---------------- layer 1 ----------------
    gat_precompute<<<1, 64, 0, stream>>>(We1, att1, emb, qbuf, etab, C1);
    gat_gemm_wmma<F1, F2><<<dim3(cdiv(N_NODES, 64), F2 / 64), 128, 0, stream>>>(acc0, W1, xw,
                                                                                N_NODES);
    gat_node_scalars<C1><<<cdiv(N_NODES * NH, B), B, 0, stream>>>(xw, att1, fillb, qbuf,
                                                                  asrc, adst, aself);
    gat_fill_u32<<<cdiv(N_NODES * NH, B), B, 0, stream>>>(menc, N_NODES * NH, 0u);
    gat_fill_f32<<<cdiv(N_NODES * NH, B), B, 0, stream>>>(ssum, N_NODES * NH, 0.f);
    // init output slice for snapshot t with bias b1; out[n][t][f]
    gat_init_out_bias<<<cdiv(N_NODES * F2, B), B, 0, stream>>>(out + (size_t)t * F2, b1,
                                                               T_SNAP * F2, F2, N_NODES * F2);
    gat_edge_logits<<<cdiv(E2, B), B, 0, stream>>>(srcP, dstP, etP, asrc, adst, aself,
                                                   etab, wexp, menc);
    gat_edge_exp<<<cdiv(E2 * NH, B), B, 0, stream>>>(dstP, wexp, menc, ssum);
    gat_scatter<C1><<<cdiv(E2 * F2, B), B, 0, stream>>>(srcP, dstP, wexp, ssum, xw,
                                                        out + (size_t)t * F2, T_SNAP * F2);
  }
}